// LogicLayer_52063593562999
// MI455X (gfx1250) — compile-verified
//
#include <hip/hip_runtime.h>
#include <hip/hip_bf16.h>

typedef __attribute__((ext_vector_type(2))) float v2f;
typedef __attribute__((ext_vector_type(8))) float v8f;

// GATE_COEFFS rows packed as nibbles: bits[4n+3:4n] = coeff_n + 2  (coeffs in -2..2)
__device__ constexpr unsigned kGatePack[16] = {
    0x2222u, 0x3222u, 0x1232u, 0x2232u,
    0x1322u, 0x2322u, 0x0332u, 0x1332u,
    0x3113u, 0x4113u, 0x2123u, 0x3123u,
    0x2213u, 0x3213u, 0x1223u, 0x2223u
};

// One wave32 per 16-neuron tile: k[tile] = softmax(W[tile]) @ G via
// 4 chained V_WMMA_F32_16X16X4_F32 (K=16 split into 4x K=4), f32-exact.
__global__ void __launch_bounds__(32)
logic_coeff_kernel(const float* __restrict__ w, float* __restrict__ k_out,
                   int out_dim)
{
    const int lane = (int)threadIdx.x;   // 0..31
    const int m    = lane & 15;          // A-row / D-col index for this lane
    const int hi   = lane >> 4;          // lane half (0/1)
    const int tile = (int)blockIdx.x;

    int row = tile * 16 + m;
    if (row >= out_dim) row = out_dim - 1;   // clamp (keep EXEC all-ones)

    // 16-float weight row, 64B aligned -> 4x b128 loads
    const float4* wr4 = reinterpret_cast<const float4*>(w + (size_t)row * 16);
    float wv[16];
    #pragma unroll
    for (int q = 0; q < 4; ++q) {
        const float4 t = wr4[q];
        wv[4*q + 0] = t.x; wv[4*q + 1] = t.y;
        wv[4*q + 2] = t.z; wv[4*q + 3] = t.w;
    }

    // exact softmax (lane pairs compute redundantly)
    float mx = wv[0];
    #pragma unroll
    for (int g = 1; g < 16; ++g) mx = fmaxf(mx, wv[g]);
    float s = 0.f;
    #pragma unroll
    for (int g = 0; g < 16; ++g) { wv[g] = expf(wv[g] - mx); s += wv[g]; }
    const float inv = 1.0f / s;
    #pragma unroll
    for (int g = 0; g < 16; ++g) wv[g] *= inv;

    const int  n   = m;          // D/B column this lane holds
    const int  sh  = (n & 3) * 4;
    const bool act = (n < 4);
    v8f acc = {};                // C/D 16x16 f32 accumulator (8 VGPRs)

    #pragma unroll
    for (int c = 0; c < 4; ++c) {
        v2f a, b;
        // A 16x4 chunk: lanes 0-15 hold K={0,1}, lanes 16-31 hold K={2,3}
        a.x = hi ? wv[c*4 + 2] : wv[c*4 + 0];
        a.y = hi ? wv[c*4 + 3] : wv[c*4 + 1];
        // B 4x16 chunk: immediates decoded from packed nibbles (no memory)
        const unsigned p0 = hi ? kGatePack[c*4 + 2] : kGatePack[c*4 + 0];
        const unsigned p1 = hi ? kGatePack[c*4 + 3] : kGatePack[c*4 + 1];
        const float g0 = (float)(int)((p0 >> sh) & 0xFu) - 2.0f;
        const float g1 = (float)(int)((p1 >> sh) & 0xFu) - 2.0f;
        b.x = act ? g0 : 0.0f;
        b.y = act ? g1 : 0.0f;
        // D = A(16x4) x B(4x16) + C
        acc = __builtin_amdgcn_wmma_f32_16x16x4_f32(
            false, a, false, b, (short)0, acc, false, false);
    }

    // D layout: acc[r] = D[r + 8*hi][n]; columns 0..3 hold k0..k3
    if (act) {
        #pragma unroll
        for (int r = 0; r < 8; ++r) {
            const int out_m = tile * 16 + r + 8 * hi;
            if (out_m < out_dim)
                k_out[(size_t)out_m * 4 + n] = acc[r];
        }
    }
}

// Memory-bound gather/eval: one thread per output column, ROWS batch rows
// per thread. 2*ROWS independent gathers in flight; NT stores keep L2 for x.
constexpr int ROWS = 8;

__global__ void __launch_bounds__(256)
logic_eval_kernel(const float* __restrict__ x, const float* __restrict__ k4,
                  const long long* __restrict__ a_idx,
                  const long long* __restrict__ b_idx,
                  float* __restrict__ out, int in_dim, int out_dim, int batch)
{
    const int j = (int)(blockIdx.x * blockDim.x + threadIdx.x);
    if (j >= out_dim) return;

    const float4    kc = reinterpret_cast<const float4*>(k4)[j];
    const long long ai = a_idx[j];
    const long long bi = b_idx[j];
    const float* xa = x + ai;
    const float* xb = x + bi;

    const int i0 = (int)blockIdx.y * ROWS;
    size_t base = (size_t)i0 * (size_t)in_dim;

    if (i0 + ROWS <= batch) {
        #pragma unroll
        for (int r = 0; r < ROWS; ++r) {
            const float a = xa[base];
            const float b = xb[base];
            const float v = fmaf(kc.w, a * b, fmaf(kc.z, b, fmaf(kc.y, a, kc.x)));
            __builtin_nontemporal_store(v, out + base + (size_t)j);
            base += (size_t)in_dim;
        }
    } else {
        for (int r = 0; r < ROWS && (i0 + r) < batch; ++r) {
            const float a = xa[base];
            const float b = xb[base];
            const float v = fmaf(kc.w, a * b, fmaf(kc.z, b, fmaf(kc.y, a, kc.x)));
            __builtin_nontemporal_store(v, out + base + (size_t)j);
            base += (size_t)in_dim;
        }
    }
}

extern "C" void kernel_launch(void* const* d_in, const int* in_sizes, int n_in,
                              void* d_out, int out_size, void* d_ws, size_t ws_size,
                              hipStream_t stream)
{
    const float*     x     = (const float*)d_in[0];
    const float*     w     = (const float*)d_in[1];
    const long long* a_idx = (const long long*)d_in[2];
    const long long* b_idx = (const long long*)d_in[3];
    float* out = (float*)d_out;
    float* k4  = (float*)d_ws;   // out_dim x 4 coefficient table (256 KB)

    const int out_dim = in_sizes[2];            // a_idx length == OUT_DIM
    const int batch   = out_size / out_dim;     // 4096
    const int in_dim  = in_sizes[0] / batch;    // 16384

    const int tiles = (out_dim + 15) / 16;
    logic_coeff_kernel<<<tiles, 32, 0, stream>>>(w, k4, out_dim);

    dim3 grid((out_dim + 255) / 256, (batch + ROWS - 1) / ROWS);
    logic_eval_kernel<<<grid, 256, 0, stream>>>(x, k4, a_idx, b_idx, out,
                                                in_dim, out_dim, batch);
}